// BHP_34205119545991
// MI455X (gfx1250) — compile-verified
//
#include <hip/hip_runtime.h>
#include <math.h>

typedef float v2f __attribute__((ext_vector_type(2)));
typedef float v8f __attribute__((ext_vector_type(8)));

#define BB   2048
#define DD   128
#define CC   8
#define PP   32
#define NTOT 4128          // 2*BB + PP, divisible by 16
#define NT   258           // NTOT / 16
#define TINV 10.0f         // 1 / temperature

// ---------------------------------------------------------------------------
// setup: zero class counters
__global__ void BHP_zero_kernel(int* cnt) {
    if (threadIdx.x < CC) cnt[threadIdx.x] = 0;
}

// setup: build feats [NTOT][DD], labels t[NTOT], class counts cnt[CC]
__global__ void BHP_build_kernel(const float* __restrict__ proxy,
                                 const float* __restrict__ features,
                                 const int*   __restrict__ targets,
                                 float* __restrict__ feats,
                                 int*   __restrict__ tArr,
                                 int*   __restrict__ cnt) {
    int g     = blockIdx.x * blockDim.x + threadIdx.x;   // NTOT*32 threads
    int row   = g >> 5;
    int chunk = g & 31;                                  // 32 float4 chunks per row
    if (row >= NTOT) return;

    const float* src;
    if (row < BB)            src = features + (size_t)row * 2 * DD;            // view 0
    else if (row < 2 * BB)   src = features + ((size_t)(row - BB) * 2 + 1) * DD; // view 1
    else                     src = proxy + (size_t)(row - 2 * BB) * DD;

    const float4* s4 = (const float4*)src;
    float4*       d4 = (float4*)(feats + (size_t)row * DD);
    d4[chunk] = s4[chunk];

    if (chunk == 0) {
        int t;
        if (row < 2 * BB) {
            t = targets[row < BB ? row : row - BB];
        } else {
            int p = row - 2 * BB;
            const int cum[CC] = {8, 14, 19, 23, 26, 29, 31, 32}; // cumsum PROXY_NUM_LIST
            t = 0;
            #pragma unroll
            for (int c = 0; c < CC; ++c) if (p >= cum[c]) t = c + 1;
        }
        tArr[row] = t;
        atomicAdd(&cnt[t], 1);
    }
}

// setup: per-column class-count weight wcol[j] = cnt[t[j]]
__global__ void BHP_wcol_kernel(const int* __restrict__ tArr,
                                const int* __restrict__ cnt,
                                float* __restrict__ wcol) {
    int j = blockIdx.x * blockDim.x + threadIdx.x;
    if (j < NTOT) wcol[j] = (float)cnt[tArr[j]];
}

// ---------------------------------------------------------------------------
// fused Gram + masked weighted log-softmax row statistics.
// One WG = one 16-row tile; 8 wave32 stride over the 258 column tiles.
__global__ __launch_bounds__(256, 2)
void BHP_gram_loss_kernel(const float* __restrict__ F,
                          const int*   __restrict__ T,
                          const float* __restrict__ W,
                          const int*   __restrict__ cnt,
                          float* __restrict__ rowv) {
    __shared__ float sA[16 * DD];   // 8 KB row block
    __shared__ float sM[16];
    __shared__ int   sT[16];
    __shared__ float redS[8][16];
    __shared__ float redA[8][16];

    const int tid     = threadIdx.x;
    const int wave    = tid >> 5;
    const int lane    = tid & 31;
    const int hi      = lane >> 4;   // which 16-lane half
    const int ln      = lane & 15;
    const int rowBase = blockIdx.x * 16;

    // stage the 16x128 row block into LDS
    for (int i = tid; i < 16 * DD; i += 256)
        sA[i] = F[(size_t)(rowBase + (i >> 7)) * DD + (i & 127)];
    if (tid < 16) sT[tid] = T[rowBase + tid];
    __syncthreads();

    // Preload A fragments per ISA 32-bit A 16x4 layout:
    // lane (ln) -> row M, half (hi) -> K pair {2hi, 2hi+1}
    v2f a[32];
    #pragma unroll
    for (int kk = 0; kk < 32; ++kk) {
        int kc = kk * 4 + hi * 2;
        a[kk] = *(const v2f*)&sA[ln * DD + kc];
    }

    // ---------------- pass 1: row max over all columns (incl. diagonal) ----
    float vmax[8];
    #pragma unroll
    for (int r = 0; r < 8; ++r) vmax[r] = -__builtin_inff();

    for (int c = wave; c < NT; c += 8) {
        const float* Bbase = F + (size_t)(c * 16 + ln) * DD + hi * 2;
        v8f acc = {};
        #pragma unroll
        for (int kk = 0; kk < 32; ++kk) {
            v2f b = *(const v2f*)(Bbase + kk * 4);
            acc = __builtin_amdgcn_wmma_f32_16x16x4_f32(false, a[kk], false, b,
                                                        (short)0, acc, false, false);
        }
        #pragma unroll
        for (int r = 0; r < 8; ++r) vmax[r] = fmaxf(vmax[r], acc[r] * TINV);
    }
    #pragma unroll
    for (int off = 8; off >= 1; off >>= 1)
        #pragma unroll
        for (int r = 0; r < 8; ++r)
            vmax[r] = fmaxf(vmax[r], __shfl_xor(vmax[r], off, 32));
    if (ln == 0)
        #pragma unroll
        for (int r = 0; r < 8; ++r) redS[wave][r + 8 * hi] = vmax[r];
    __syncthreads();
    if (tid < 16) {
        float m = redS[0][tid];
        #pragma unroll
        for (int w = 1; w < 8; ++w) m = fmaxf(m, redS[w][tid]);
        sM[tid] = m;
    }
    __syncthreads();

    float myM[8]; int myT[8];
    #pragma unroll
    for (int r = 0; r < 8; ++r) { myM[r] = sM[r + 8 * hi]; myT[r] = sT[r + 8 * hi]; }

    // ---------------- pass 2: weighted exp-sum + positive log-prob sum -----
    float S[8], Aw[8];
    #pragma unroll
    for (int r = 0; r < 8; ++r) { S[r] = 0.f; Aw[r] = 0.f; }

    for (int c = wave; c < NT; c += 8) {
        const int col = c * 16 + ln;
        const int   tc = T[col];
        const float wc = W[col];
        const float* Bbase = F + (size_t)col * DD + hi * 2;
        v8f acc = {};
        #pragma unroll
        for (int kk = 0; kk < 32; ++kk) {
            v2f b = *(const v2f*)(Bbase + kk * 4);
            acc = __builtin_amdgcn_wmma_f32_16x16x4_f32(false, a[kk], false, b,
                                                        (short)0, acc, false, false);
        }
        #pragma unroll
        for (int r = 0; r < 8; ++r) {
            int   row  = rowBase + r + 8 * hi;
            float l    = acc[r] * TINV - myM[r];
            bool  diag = (row == col);
            float mk   = (!diag && (myT[r] == tc)) ? 1.0f : 0.0f;
            float e    = diag ? 0.0f : __expf(l);
            S[r]  += e / (wc - mk);        // per_ins_weight = cnt[t_j] - mask
            Aw[r] += mk * l;               // sum of mask*(logit - max)
        }
    }
    #pragma unroll
    for (int off = 8; off >= 1; off >>= 1)
        #pragma unroll
        for (int r = 0; r < 8; ++r) {
            S[r]  += __shfl_xor(S[r],  off, 32);
            Aw[r] += __shfl_xor(Aw[r], off, 32);
        }
    if (ln == 0)
        #pragma unroll
        for (int r = 0; r < 8; ++r) {
            redS[wave][r + 8 * hi] = S[r];
            redA[wave][r + 8 * hi] = Aw[r];
        }
    __syncthreads();
    if (tid < 16) {
        float Ss = 0.f, As = 0.f;
        #pragma unroll
        for (int w = 0; w < 8; ++w) { Ss += redS[w][tid]; As += redA[w][tid]; }
        float K = (float)cnt[sT[tid]] - 1.0f;        // mask.sum(axis=1)
        rowv[rowBase + tid] = (As - K * __logf(Ss)) / K;
    }
}

// deterministic final reduction: loss = -mean(rowv)
__global__ void BHP_final_kernel(const float* __restrict__ rowv, float* __restrict__ out) {
    __shared__ float s[256];
    float p = 0.f;
    for (int i = threadIdx.x; i < NTOT; i += 256) p += rowv[i];
    s[threadIdx.x] = p;
    __syncthreads();
    for (int off = 128; off >= 1; off >>= 1) {
        if (threadIdx.x < off) s[threadIdx.x] += s[threadIdx.x + off];
        __syncthreads();
    }
    if (threadIdx.x == 0) out[0] = -s[0] / (float)NTOT;
}

// ---------------------------------------------------------------------------
extern "C" void kernel_launch(void* const* d_in, const int* in_sizes, int n_in,
                              void* d_out, int out_size, void* d_ws, size_t ws_size,
                              hipStream_t stream) {
    const float* proxy    = (const float*)d_in[0];  // [P, D]
    const float* features = (const float*)d_in[1];  // [B, 2, D]
    const int*   targets  = (const int*)d_in[2];    // [B]
    float* out = (float*)d_out;

    // workspace layout
    float* feats = (float*)d_ws;                              // NTOT*DD f32
    int*   tArr  = (int*)(feats + (size_t)NTOT * DD);         // NTOT i32
    float* wcol  = (float*)(tArr + NTOT);                     // NTOT f32
    float* rowv  = wcol + NTOT;                               // NTOT f32
    int*   cnt   = (int*)(rowv + NTOT);                       // CC i32

    BHP_zero_kernel<<<1, 64, 0, stream>>>(cnt);
    BHP_build_kernel<<<(NTOT * 32) / 256, 256, 0, stream>>>(proxy, features, targets,
                                                            feats, tArr, cnt);
    BHP_wcol_kernel<<<(NTOT + 255) / 256, 256, 0, stream>>>(tArr, cnt, wcol);
    BHP_gram_loss_kernel<<<NT, 256, 0, stream>>>(feats, tArr, wcol, cnt, rowv);
    BHP_final_kernel<<<1, 256, 0, stream>>>(rowv, out);
}